// BreakthroughSNN_33071248179950
// MI455X (gfx1250) — compile-verified
//
#include <hip/hip_runtime.h>

typedef __attribute__((ext_vector_type(2))) float v2f;
typedef __attribute__((ext_vector_type(4))) float v4f;
typedef __attribute__((ext_vector_type(8))) float v8f;

#define SNN_B 4
#define SNN_T 16
#define SNN_S 256
#define SNN_D 512
#define SNN_N 8
#define GEMM_M (SNN_B * SNN_S)   // 1024
#define GEMM_K SNN_D             // 512
#define GEMM_N (SNN_D * SNN_N)   // 4096
#define TWO_PI_F 6.28318530717958647692f

// Fast sigmoid: v_exp_f32 + v_rcp_f32 (avoids the IEEE div_scale/fma chain).
__device__ __forceinline__ float fast_sigmoid(float x) {
  return __builtin_amdgcn_rcpf(1.0f + __expf(-x));
}

// ---------------------------------------------------------------------------
// Kernel 0: one wave computes softmax(enc_weights[4]) into ws[0..3].
// ---------------------------------------------------------------------------
__global__ void snn_prep(const float* __restrict__ encw, float* __restrict__ w) {
  if (threadIdx.x == 0) {
    float w0 = encw[0], w1 = encw[1], w2 = encw[2], w3 = encw[3];
    const float m = fmaxf(fmaxf(w0, w1), fmaxf(w2, w3));
    const float e0 = __expf(w0 - m), e1 = __expf(w1 - m);
    const float e2 = __expf(w2 - m), e3 = __expf(w3 - m);
    const float inv = __builtin_amdgcn_rcpf(e0 + e1 + e2 + e3);
    w[0] = e0 * inv; w[1] = e1 * inv; w[2] = e2 * inv; w[3] = e3 * inv;
  }
}

// ---------------------------------------------------------------------------
// Kernel 1: pop_rates[row, col] = sigmoid( emb[row,:] @ pop_W[:,col] + b[col] )
// row = b*S+s (1024 rows), col = d*N+n (4096 cols).
// One wave32 computes one 16x16 f32 tile via V_WMMA_F32_16X16X4_F32,
// 128 WMMA steps along K=512. 8 waves/block = 8 adjacent N-tiles (share A rows
// through L2). Grid = (M/16) * (N/128) = 64 * 32 = 2048 blocks.
// ---------------------------------------------------------------------------
__global__ __launch_bounds__(256) void snn_pop_gemm(
    const float* __restrict__ A,      // [1024, 512] row-major (embeddings)
    const float* __restrict__ W,      // [512, 4096] row-major (pop_W)
    const float* __restrict__ bias,   // [4096]
    float* __restrict__ rates) {      // [1024, 4096] out (sigmoid applied)
  const int lane   = threadIdx.x & 31;
  const int wave   = threadIdx.x >> 5;
  const int lane16 = lane & 15;
  const int half   = lane >> 4;          // 0: lanes 0-15, 1: lanes 16-31

  const int tilesM   = GEMM_M / 16;                    // 64
  const int tileRow  = blockIdx.x % tilesM;
  const int colGroup = blockIdx.x / tilesM;            // 0..31
  const int colBase  = (colGroup * 8 + wave) * 16;

  // A-matrix 16x4 layout: lane l<16 holds A[M=l][K=0,1] in v0,v1;
  // lane l>=16 holds A[M=l-16][K=2,3].  => per-lane: K = 2*half + {0,1}
  const float* aRow = A + (size_t)(tileRow * 16 + lane16) * GEMM_K + 2 * half;
  // B-matrix 4x16 layout mirrors C/D striping: v0 holds K=2*half, v1 K=2*half+1,
  // N = lane16 within each VGPR.
  const float* wCol = W + (size_t)(2 * half) * GEMM_N + colBase + lane16;

  v8f acc = {};
#pragma unroll 4
  for (int k = 0; k < GEMM_K; k += 4) {
    v2f a, b;
    a.x = aRow[k + 0];
    a.y = aRow[k + 1];
    b.x = wCol[(size_t)k * GEMM_N];
    b.y = wCol[(size_t)(k + 1) * GEMM_N];
    // (neg_a, A, neg_b, B, c_mod, C, reuse_a, reuse_b)
    acc = __builtin_amdgcn_wmma_f32_16x16x4_f32(false, a, false, b,
                                                (short)0, acc, false, false);
  }

  // C/D layout: VGPR r holds row M = r + 8*half, col N = lane16.
  const int col = colBase + lane16;
  const float bcol = bias[col];
#pragma unroll
  for (int r = 0; r < 8; ++r) {
    const int row = tileRow * 16 + r + 8 * half;
    rates[(size_t)row * GEMM_N + col] = fast_sigmoid(acc[r] + bcol);
  }
}

// ---------------------------------------------------------------------------
// Kernel 2: streaming combine over [B,T,S,D].  Bandwidth-bound: 256MB pop_rand
// + 32MB rate_rand in, 32MB out; pop_rates (16MB) re-read from L2.
// ---------------------------------------------------------------------------
__global__ __launch_bounds__(256) void snn_combine(
    const float* __restrict__ emb,     // [B,S,D]
    const float* __restrict__ freq,    // [D]
    const float* __restrict__ wvec,    // [4] softmax'd encoder weights (ws)
    const float* __restrict__ noise,   // [B,S,D]
    const float* __restrict__ rrand,   // [B,T,S,D]
    const float* __restrict__ prand,   // [B,T,S,D,N]
    const float* __restrict__ prates,  // [B*S, D*N] from kernel 1 (ws)
    float* __restrict__ out) {         // [B,T,S,D]
  const int i = blockIdx.x * blockDim.x + threadIdx.x;   // < B*T*S*D
  const int d  = i & (SNN_D - 1);
  const int s  = (i >> 9) & (SNN_S - 1);
  const int ts = i >> 17;                 // b*T + t
  const int t  = ts & (SNN_T - 1);
  const int b  = ts >> 4;

  // precomputed softmax weights (uniform address -> scalar-cache friendly)
  const float w0 = wvec[0], w1 = wvec[1], w2 = wvec[2], w3 = wvec[3];

  const size_t bsd = ((size_t)(b * SNN_S + s)) * SNN_D + d;
  const float e   = emb[bsd];
  const float sig = fast_sigmoid(e);

  // --- rate encoder ---
  float rate = sig * 0.9f + 0.05f + noise[bsd] * 0.1f;
  rate = fminf(fmaxf(rate, 0.0f), 1.0f);
  const float rate_spike = (rrand[i] < rate) ? 1.0f : 0.0f;

  // --- temporal encoder ---
  const int st = (int)(sig * (float)(SNN_T - 1));
  const float temporal_spike = (st == t) ? 1.0f : 0.0f;

  // --- population encoder: 8 Bernoulli draws, mean ---
  const size_t rbase = ((size_t)(b * SNN_S + s)) * GEMM_N + (size_t)d * SNN_N;
  const size_t pbase = (size_t)i * SNN_N;
  const v4f r0 = *(const v4f*)(prates + rbase);
  const v4f r1 = *(const v4f*)(prates + rbase + 4);
  const v4f x0 = *(const v4f*)(prand + pbase);
  const v4f x1 = *(const v4f*)(prand + pbase + 4);
  float cnt = 0.0f;
  cnt += (x0.x < r0.x) ? 1.0f : 0.0f;
  cnt += (x0.y < r0.y) ? 1.0f : 0.0f;
  cnt += (x0.z < r0.z) ? 1.0f : 0.0f;
  cnt += (x0.w < r0.w) ? 1.0f : 0.0f;
  cnt += (x1.x < r1.x) ? 1.0f : 0.0f;
  cnt += (x1.y < r1.y) ? 1.0f : 0.0f;
  cnt += (x1.z < r1.z) ? 1.0f : 0.0f;
  cnt += (x1.w < r1.w) ? 1.0f : 0.0f;
  const float pop_mean = cnt * 0.125f;

  // --- phase encoder ---
  const float tv = (float)t * (TWO_PI_F / (float)(SNN_T - 1));
  const float wavev = __sinf(freq[d] * tv + sig * TWO_PI_F);
  const float phase_spike = (wavev > 0.5f) ? 1.0f : 0.0f;

  out[i] = w0 * rate_spike + w1 * temporal_spike + w2 * pop_mean + w3 * phase_spike;
}

extern "C" void kernel_launch(void* const* d_in, const int* in_sizes, int n_in,
                              void* d_out, int out_size, void* d_ws, size_t ws_size,
                              hipStream_t stream) {
  (void)in_sizes; (void)n_in; (void)out_size; (void)ws_size;
  const float* emb   = (const float*)d_in[0];  // embeddings  [B,S,D]
  const float* popW  = (const float*)d_in[1];  // pop_W       [D, D*N]
  const float* popb  = (const float*)d_in[2];  // pop_b       [D*N]
  const float* freq  = (const float*)d_in[3];  // freq_bands  [D]
  const float* encw  = (const float*)d_in[4];  // enc_weights [4]
  const float* noise = (const float*)d_in[5];  // rate_noise  [B,S,D]
  const float* rrand = (const float*)d_in[6];  // rate_rand   [B,T,S,D]
  const float* prand = (const float*)d_in[7];  // pop_rand    [B,T,S,D,N]
  float* out = (float*)d_out;

  // Workspace layout: [0..3] softmax weights, [64..] pop_rates (16 MiB),
  // total requirement: 16 MiB + 256 B.
  float* wsf    = (float*)d_ws;
  float* wvec   = wsf;
  float* prates = wsf + 64;

  // Kernel 0: softmax of the 4 encoder weights (one wave).
  snn_prep<<<1, 32, 0, stream>>>(encw, wvec);

  // Kernel 1: WMMA GEMM + sigmoid into workspace.
  const int gemmBlocks = (GEMM_M / 16) * (GEMM_N / (16 * 8));  // 64 * 32
  snn_pop_gemm<<<gemmBlocks, 256, 0, stream>>>(emb, popW, popb, prates);

  // Kernel 2: streaming combine (same stream -> ordered after 0 and 1).
  const int total = SNN_B * SNN_T * SNN_S * SNN_D;             // 8,388,608
  snn_combine<<<total / 256, 256, 0, stream>>>(emb, freq, wvec, noise,
                                               rrand, prand, prates, out);
}